// StructNDeconv2D_74294344286413
// MI455X (gfx1250) — compile-verified
//
#include <hip/hip_runtime.h>
#include <hip/hip_bf16.h>
#include <stdint.h>

// StructNDeconv2D (normalized depthwise conv_transpose2d, K=3, stride=2, pad=1)
// Memory-bound: ~668 MB traffic -> ~29us floor @ 23.3 TB/s.
// Round-3 restructure: softplus hoisted to a one-shot prep kernel (the inlined
// scalarized expf/log1pf was 1241 SALU ops per wave per block!), and the main
// kernel now processes 16 output rows x 511 cols per block with 9 input rows
// staged in LDS via CDNA5 async global->LDS loads. Each thread owns an
// (even,odd) column pair -> no parity divergence; register-carried row pipe.

#define B_   8
#define C_   32
#define H_   256
#define W_   256
#define OH_  511
#define OW_  511
#define HW_IN ((size_t)H_ * W_)
#define OHW_  ((size_t)OH_ * OW_)

// ---------- CDNA5 async global->LDS helpers ----------
typedef __attribute__((address_space(1))) int g_i32_t;
typedef __attribute__((address_space(3))) int l_i32_t;

__device__ __forceinline__ void async_g2l_b32(const float* gsrc, void* lds_dst) {
#if __has_builtin(__builtin_amdgcn_global_load_async_to_lds_b32)
  __builtin_amdgcn_global_load_async_to_lds_b32(
      (g_i32_t*)gsrc, (l_i32_t*)lds_dst, 0, 0);
#else
  uint32_t lds_off = (uint32_t)(uintptr_t)lds_dst;
  asm volatile("global_load_async_to_lds_b32 %0, %1, off"
               :: "v"(lds_off), "v"(gsrc)
               : "memory");
#endif
}

__device__ __forceinline__ void wait_async0() {
#if __has_builtin(__builtin_amdgcn_s_wait_asynccnt)
  __builtin_amdgcn_s_wait_asynccnt(0);
#else
  asm volatile("s_wait_asynccnt 0" ::: "memory");
#endif
}

// ---------- one-shot channel-parameter prep (32 threads total) ----------
// params[c*16 + 0..8]  = softplus(weights[c])
// params[c*16 + 9..12] = 1/(cdenom+eps) for parities (ee, eo, oe, oo)
// params[c*16 + 13]    = bias[c]
__global__ void StructNDeconv2D_prep(const float* __restrict__ sw,
                                     const float* __restrict__ bias,
                                     float* __restrict__ params) {
  const int c = threadIdx.x;
  if (c >= C_) return;
  float w[9];
#pragma unroll
  for (int i = 0; i < 9; ++i) {
    float x  = sw[c * 9 + i];
    float ax = fabsf(x);
    w[i] = fmaxf(x, 0.0f) + log1pf(expf(-ax));   // stable softplus
    params[c * 16 + i] = w[i];
  }
  const float eps = 1e-20f;
  params[c * 16 +  9] = 1.0f / (w[4] + eps);                        // oy even, ox even
  params[c * 16 + 10] = 1.0f / (w[3] + w[5] + eps);                 // oy even, ox odd
  params[c * 16 + 11] = 1.0f / (w[1] + w[7] + eps);                 // oy odd,  ox even
  params[c * 16 + 12] = 1.0f / (w[0] + w[2] + w[6] + w[8] + eps);   // oy odd,  ox odd
  params[c * 16 + 13] = bias[c];
  params[c * 16 + 14] = 0.0f;
  params[c * 16 + 15] = 0.0f;
}

// ---------- main kernel ----------
// grid = (32 row-tiles, 256 images); block = 256 threads (8 wave32).
// Block: output rows [16*bx, 16*bx+15] x all 511 cols of image z.
// Thread t: output columns (2t, 2t+1), walking 8 row pairs.
__global__ __launch_bounds__(256) void StructNDeconv2D_main(
    const float* __restrict__ d, const float* __restrict__ cd,
    const float* __restrict__ params,
    float* __restrict__ dout, float* __restrict__ cdout)
{
  __shared__ float s_d [9][260];
  __shared__ float s_cd[9][260];

  const int tid = threadIdx.x;
  const int oy0 = blockIdx.x << 4;
  const int z   = blockIdx.y;          // b*C + c
  const int c   = z & (C_ - 1);
  const int m0  = oy0 >> 1;            // first input row needed

  const size_t in_base = (size_t)z * HW_IN;

  // Stage 9 full input rows of d and cd (256 lanes x b32 each, async).
#pragma unroll
  for (int r = 0; r < 9; ++r) {
    int iy = m0 + r; if (iy > H_ - 1) iy = H_ - 1;   // clamp: row 8 of last tile is never read
    const size_t g = in_base + (size_t)iy * W_ + tid;
    async_g2l_b32(d  + g, &s_d [r][tid]);
    async_g2l_b32(cd + g, &s_cd[r][tid]);
  }

  // Uniform scalar loads of precomputed channel params (overlaps async copy).
  const float* P = params + c * 16;
  const float w0 = P[0], w1 = P[1], w2 = P[2],
              w3 = P[3], w4 = P[4], w5 = P[5],
              w6 = P[6], w7 = P[7], w8 = P[8];
  const float icd_ee = P[9], icd_eo = P[10], icd_oe = P[11], icd_oo = P[12];
  const float bc = P[13];

  wait_async0();
  __syncthreads();

  const int  t   = tid;
  const int  tp1 = (t < 255) ? t + 1 : 255;          // clamped; lane 255 odd-col is discarded
  const bool colOddValid = (t < 255);                // ox=2t+1 <= 509

  float* pd = dout  + (size_t)z * OHW_ + (size_t)oy0 * OW_ + (t << 1);
  float* pc = cdout + (size_t)z * OHW_ + (size_t)oy0 * OW_ + (t << 1);

  // Register-carried row pipeline: (a,p) = (cd, d) at cols t, t+1 of row j.
  float a0 = s_cd[0][t],   p0 = s_d[0][t];
  float a1 = s_cd[0][tp1], p1 = s_d[0][tp1];
  float t0 = a0 * p0, t1v = a1 * p1;                 // cd*d products

#pragma unroll
  for (int j = 0; j < 8; ++j) {
    float b0 = s_cd[j + 1][t],   q0 = s_d[j + 1][t];
    float b1 = s_cd[j + 1][tp1], q1 = s_d[j + 1][tp1];
    float u0 = b0 * q0, u1 = b1 * q1;

    // even output row oy0+2j: ky=1 taps on input row j
    float nom_ee = w4 * t0;
    float den_ee = w4 * a0;
    float nom_eo = fmaf(w3, t1v, w5 * t0);
    float den_eo = fmaf(w3, a1,  w5 * a0);
    // odd output row oy0+2j+1: ky=2 on row j, ky=0 on row j+1
    float nom_oe = fmaf(w1, u0, w7 * t0);
    float den_oe = fmaf(w1, b0, w7 * a0);
    float nom_oo = fmaf(w0, u1, fmaf(w2, u0, fmaf(w6, t1v, w8 * t0)));
    float den_oo = fmaf(w0, b1, fmaf(w2, b0, fmaf(w6, a1, w8 * a0)));

    float dee = fmaf(nom_ee, __builtin_amdgcn_rcpf(den_ee + 1e-20f), bc);
    float deo = fmaf(nom_eo, __builtin_amdgcn_rcpf(den_eo + 1e-20f), bc);
    float doe = fmaf(nom_oe, __builtin_amdgcn_rcpf(den_oe + 1e-20f), bc);
    float doo = fmaf(nom_oo, __builtin_amdgcn_rcpf(den_oo + 1e-20f), bc);
    float cee = den_ee * icd_ee;
    float ceo = den_eo * icd_eo;
    float coe = den_oe * icd_oe;
    float coo = den_oo * icd_oo;

    const int  off = j * (2 * OW_);                  // immediate offsets in the unrolled body
    const bool rowOddValid = (oy0 + 2 * j + 1) < OH_;
    pd[off] = dee;  pc[off] = cee;
    if (colOddValid) { pd[off + 1] = deo;  pc[off + 1] = ceo; }
    if (rowOddValid) {
      pd[off + OW_] = doe;  pc[off + OW_] = coe;
      if (colOddValid) { pd[off + OW_ + 1] = doo;  pc[off + OW_ + 1] = coo; }
    }

    a0 = b0; p0 = q0; a1 = b1; p1 = q1; t0 = u0; t1v = u1;
  }
}

extern "C" void kernel_launch(void* const* d_in, const int* in_sizes, int n_in,
                              void* d_out, int out_size, void* d_ws, size_t ws_size,
                              hipStream_t stream) {
  (void)in_sizes; (void)n_in; (void)out_size; (void)ws_size;
  const float* d    = (const float*)d_in[0];   // [8,32,256,256]
  const float* cd   = (const float*)d_in[1];   // [8,32,256,256]
  const float* sw   = (const float*)d_in[2];   // [32,1,3,3]
  const float* bias = (const float*)d_in[3];   // [1,32,1,1]
  float* params = (float*)d_ws;                // 32*16 floats = 2 KB scratch
  float* dout   = (float*)d_out;
  float* cdout  = dout + (size_t)B_ * C_ * OH_ * OW_;

  StructNDeconv2D_prep<<<1, 32, 0, stream>>>(sw, bias, params);

  dim3 grid(32, B_ * C_);   // 32 row-tiles of 16 rows, 256 images
  dim3 block(256);
  StructNDeconv2D_main<<<grid, block, 0, stream>>>(d, cd, params, dout, cdout);
}